// GATNet_53455162966194
// MI455X (gfx1250) — compile-verified
//
#include <hip/hip_runtime.h>
#include <hip/hip_bf16.h>
#include <math.h>

// ---------------------------------------------------------------------------
// GAT 2-layer network for MI455X (gfx1250).
//  * Edge phases: atomic-based segment max / segment sum (L2-resident, 192MB L2
//    holds the whole ~115MB working set), with global_prefetch on the random
//    source-feature gather.
//  * Layer-2 GEMM [N,64]@[64,128] in full fp32 via V_WMMA_F32_16X16X4_F32:
//    W2 staged in LDS (32KB of the 320KB/WGP), one wave per 16x128 strip,
//    fully unrolled 128 WMMAs per wave.
// ---------------------------------------------------------------------------

typedef __attribute__((ext_vector_type(2))) float v2f;
typedef __attribute__((ext_vector_type(8))) float v8f;

#define NEG_SLOPE 0.2f
#define N_GRAPHS 64
#define ORD_NEG_INF 0x007FFFFFu   // f2ord(-inf)

__device__ __forceinline__ unsigned f2ord(float f) {
    unsigned u = __float_as_uint(f);
    return (u & 0x80000000u) ? ~u : (u | 0x80000000u);
}
__device__ __forceinline__ float ord2f(unsigned u) {
    u = (u & 0x80000000u) ? (u & 0x7FFFFFFFu) : ~u;
    return __uint_as_float(u);
}
__device__ __forceinline__ float lrelu(float v) { return v > 0.f ? v : NEG_SLOPE * v; }

// --------------------------------------------------------------------------
// zero small pooled/count region
__global__ void k_zero(float* __restrict__ p, int count) {
    int i = blockIdx.x * blockDim.x + threadIdx.x;
    if (i < count) p[i] = 0.f;
}

// --------------------------------------------------------------------------
// Layer 1 features (rank-1): h1 = x * W1, attention logits, init accumulators.
__global__ void k_l1_feats(const float* __restrict__ x, const float* __restrict__ W1,
                           const float* __restrict__ as1, const float* __restrict__ ad1,
                           float* __restrict__ h1, float* __restrict__ asrc,
                           float* __restrict__ adst, unsigned* __restrict__ mord,
                           float* __restrict__ denom, float* __restrict__ out1, int n) {
    int i = blockIdx.x * blockDim.x + threadIdx.x;
    if (i >= n) return;
    float xv = x[i];
#pragma unroll
    for (int h = 0; h < 8; ++h) {
        float sa = 0.f, sd = 0.f;
#pragma unroll
        for (int c = 0; c < 8; ++c) {
            float f = xv * W1[h * 8 + c];
            h1[(size_t)i * 64 + h * 8 + c] = f;
            out1[(size_t)i * 64 + h * 8 + c] = 0.f;
            sa += f * as1[h * 8 + c];
            sd += f * ad1[h * 8 + c];
        }
        asrc[(size_t)i * 8 + h] = sa;
        adst[(size_t)i * 8 + h] = sd;
        mord[(size_t)i * 8 + h] = ORD_NEG_INF;
        denom[(size_t)i * 8 + h] = 0.f;
    }
}

// --------------------------------------------------------------------------
// Edge pass 1: segment max of leaky_relu(a_src[s]+a_dst[d]) per (dst, head).
template <int H>
__global__ void k_edge_max(const int* __restrict__ src, const int* __restrict__ dst,
                           int n_edges, int n_nodes,
                           const float* __restrict__ asrc, const float* __restrict__ adst,
                           unsigned* __restrict__ mord) {
    long long idx = (long long)blockIdx.x * blockDim.x + threadIdx.x;
    long long EH = (long long)(n_edges + n_nodes) * H;
    if (idx >= EH) return;
    int e = (int)(idx / H);
    int h = (int)(idx - (long long)e * H);
    int s, d;
    if (e < n_edges) { s = src[e]; d = dst[e]; } else { s = d = e - n_edges; }
    float v = lrelu(asrc[(size_t)s * H + h] + adst[(size_t)d * H + h]);
    atomicMax(&mord[(size_t)d * H + h], f2ord(v));
}

// --------------------------------------------------------------------------
// Edge pass 2: ex = exp(e - m[d]); denom[d] += ex; out[d] += ex * feat[s].
// (softmax folded: out/denom at the node pass == softmax-weighted sum)
template <int H, int C>
__global__ void k_edge_acc(const int* __restrict__ src, const int* __restrict__ dst,
                           int n_edges, int n_nodes,
                           const float* __restrict__ asrc, const float* __restrict__ adst,
                           const unsigned* __restrict__ mord,
                           const float* __restrict__ feat,
                           float* __restrict__ denom, float* __restrict__ outacc) {
    long long idx = (long long)blockIdx.x * blockDim.x + threadIdx.x;
    long long EH = (long long)(n_edges + n_nodes) * H;
    if (idx >= EH) return;
    int e = (int)(idx / H);
    int h = (int)(idx - (long long)e * H);
    int s, d;
    if (e < n_edges) { s = src[e]; d = dst[e]; } else { s = d = e - n_edges; }
    const float* fs = feat + ((size_t)s * H + h) * C;
    __builtin_prefetch(fs, 0, 3);   // global_prefetch: start random row fetch early
    float v = lrelu(asrc[(size_t)s * H + h] + adst[(size_t)d * H + h]);
    float m = ord2f(mord[(size_t)d * H + h]);
    float ex = __expf(v - m);
    atomicAdd(&denom[(size_t)d * H + h], ex);
    float* od = outacc + ((size_t)d * H + h) * C;
#pragma unroll
    for (int c = 0; c < C; ++c) atomicAdd(&od[c], ex * fs[c]);
}

// --------------------------------------------------------------------------
// Layer 1 finalize: normalize by denom, add bias, ELU (in place -> h1_act).
__global__ void k_fin1(float* __restrict__ out1, const float* __restrict__ denom,
                       const float* __restrict__ b1, int n) {
    long long idx = (long long)blockIdx.x * blockDim.x + threadIdx.x;
    if (idx >= (long long)n * 64) return;
    int i = (int)(idx >> 6);
    int j = (int)(idx & 63);
    int h = j >> 3;
    float v = out1[idx] / denom[(size_t)i * 8 + h] + b1[j];
    out1[idx] = v > 0.f ? v : (__expf(v) - 1.f);
}

// --------------------------------------------------------------------------
// fp32 WMMA GEMM: D[M,NC] = A[M,K] @ B[K,NC], K%4==0, NC%16==0, compile-time K/NC.
// B staged in LDS. One wave32 per 16-row strip computing all NC/16 tiles:
// A fragment loaded once per k-step, 128 fully unrolled V_WMMA_F32_16X16X4_F32.
// A frag (16x4): lanes 0-15 row=lane, K pair {k,k+1}; lanes 16-31 K pair {k+2,k+3}.
// B frag (4x16): vgpr0 = rows {k,k+2} striped across lane halves; vgpr1 = {k+1,k+3}.
// C/D (16x16): vgpr v -> rows v (lanes 0-15) and v+8 (lanes 16-31), col = lane&15.
template <int K, int NC>
__global__ void k_gemm_wmma_f32(const float* __restrict__ A, const float* __restrict__ B,
                                float* __restrict__ D, int M) {
    __shared__ float Bs[K * NC];                    // 64*128*4B = 32KB LDS
    for (int t = threadIdx.x; t < K * NC; t += blockDim.x) Bs[t] = B[t];
    __syncthreads();

    const int lane = threadIdx.x & 31;
    const int wave = threadIdx.x >> 5;
    int strip = blockIdx.x * (blockDim.x >> 5) + wave;
    if (strip * 16 >= M) return;                    // wave-uniform: EXEC all-1s at WMMA

    const int r = lane & 15;
    const int hi = lane >> 4;                       // selects K sub-pair {0,1} vs {2,3}
    int arow = strip * 16 + r;
    if (arow >= M) arow = M - 1;                    // clamp (harmless duplicate read)
    const float* Ap = A + (size_t)arow * K;

    const v8f vzero = {0.f, 0.f, 0.f, 0.f, 0.f, 0.f, 0.f, 0.f};
    v8f acc[NC / 16];
#pragma unroll
    for (int nt = 0; nt < NC / 16; ++nt) acc[nt] = vzero;

#pragma unroll
    for (int k = 0; k < K; k += 4) {
        const int kk = k + hi * 2;
        v2f a;
        a.x = Ap[kk];
        a.y = Ap[kk + 1];
#pragma unroll
        for (int nt = 0; nt < NC / 16; ++nt) {
            v2f b;
            b.x = Bs[kk * NC + nt * 16 + r];
            b.y = Bs[(kk + 1) * NC + nt * 16 + r];
            acc[nt] = __builtin_amdgcn_wmma_f32_16x16x4_f32(
                /*neg_a=*/false, a, /*neg_b=*/false, b,
                /*c_mod=*/(short)0, acc[nt], /*reuse_a=*/false, /*reuse_b=*/false);
        }
    }

#pragma unroll
    for (int nt = 0; nt < NC / 16; ++nt) {
#pragma unroll
        for (int v = 0; v < 8; ++v) {
            int rr = strip * 16 + v + hi * 8;
            if (rr < M) D[(size_t)rr * NC + nt * 16 + r] = acc[nt][v];
        }
    }
}

// --------------------------------------------------------------------------
// Layer 2 attention logits from h2, and re-init of m/denom/out2 accumulators.
__global__ void k_l2_att(const float* __restrict__ h2, const float* __restrict__ as2,
                         const float* __restrict__ ad2, float* __restrict__ asrc,
                         float* __restrict__ adst, unsigned* __restrict__ mord,
                         float* __restrict__ denom, float* __restrict__ out2, int n) {
    int i = blockIdx.x * blockDim.x + threadIdx.x;
    if (i >= n) return;
#pragma unroll
    for (int h = 0; h < 8; ++h) {
        float sa = 0.f, sd = 0.f;
#pragma unroll
        for (int c = 0; c < 16; ++c) {
            float f = h2[(size_t)i * 128 + h * 16 + c];
            out2[(size_t)i * 128 + h * 16 + c] = 0.f;
            sa += f * as2[h * 16 + c];
            sd += f * ad2[h * 16 + c];
        }
        asrc[(size_t)i * 8 + h] = sa;
        adst[(size_t)i * 8 + h] = sd;
        mord[(size_t)i * 8 + h] = ORD_NEG_INF;
        denom[(size_t)i * 8 + h] = 0.f;
    }
}

// --------------------------------------------------------------------------
// Layer 2 finalize: normalize, mean over heads, +b2, then global mean-pool
// accumulation per graph (sums + counts).
__global__ void k_fin2(const float* __restrict__ out2, const float* __restrict__ denom,
                       const float* __restrict__ b2, const int* __restrict__ batch,
                       float* __restrict__ pooled, float* __restrict__ cnt, int n) {
    int i = blockIdx.x * blockDim.x + threadIdx.x;
    if (i >= n) return;
    int g = batch[i];
    float inv[8];
#pragma unroll
    for (int h = 0; h < 8; ++h) inv[h] = 1.f / denom[(size_t)i * 8 + h];
#pragma unroll
    for (int c = 0; c < 16; ++c) {
        float s = 0.f;
#pragma unroll
        for (int h = 0; h < 8; ++h) s += out2[(size_t)i * 128 + h * 16 + c] * inv[h];
        atomicAdd(&pooled[g * 16 + c], s * 0.125f + b2[c]);
    }
    atomicAdd(&cnt[g], 1.f);
}

// --------------------------------------------------------------------------
// Final FC: out[g,o] = (pooled[g,:]/cnt[g]) @ Wfc + bfc.  64x4 = 256 threads.
__global__ void k_fc(const float* __restrict__ pooled, const float* __restrict__ cnt,
                     const float* __restrict__ Wfc, const float* __restrict__ bfc,
                     float* __restrict__ out) {
    int t = threadIdx.x;
    if (t >= N_GRAPHS * 4) return;
    int g = t >> 2, o = t & 3;
    float c = cnt[g];
    c = c > 1.f ? c : 1.f;
    float s = 0.f;
#pragma unroll
    for (int k = 0; k < 16; ++k) s += (pooled[g * 16 + k] / c) * Wfc[k * 4 + o];
    out[t] = s + bfc[o];
}

// ---------------------------------------------------------------------------
extern "C" void kernel_launch(void* const* d_in, const int* in_sizes, int n_in,
                              void* d_out, int out_size, void* d_ws, size_t ws_size,
                              hipStream_t stream) {
    const float* x    = (const float*)d_in[0];
    const float* W1   = (const float*)d_in[1];
    const float* as1  = (const float*)d_in[2];
    const float* ad1  = (const float*)d_in[3];
    const float* b1   = (const float*)d_in[4];
    const float* W2   = (const float*)d_in[5];
    const float* as2  = (const float*)d_in[6];
    const float* ad2  = (const float*)d_in[7];
    const float* b2   = (const float*)d_in[8];
    const float* Wfc  = (const float*)d_in[9];
    const float* bfc  = (const float*)d_in[10];
    const int*   ei   = (const int*)d_in[11];
    const int*   batch= (const int*)d_in[12];

    const int n  = in_sizes[0];        // 100000 nodes
    const int ne = in_sizes[11] / 2;   // 1600000 edges (self loops added on the fly)
    const int* src = ei;
    const int* dst = ei + ne;

    // Workspace layout (floats). out2 reuses the h1/out1 region (dead by then).
    float* ws      = (float*)d_ws;
    float*    h1    = ws;                                // N*64   (layer1 pre-features)
    float*    out1  = h1 + (size_t)n * 64;               // N*64   (layer1 accum -> h1_act)
    float*    h2    = out1 + (size_t)n * 64;             // N*128  (layer2 pre-features)
    float*    asrc  = h2 + (size_t)n * 128;              // N*8
    float*    adst  = asrc + (size_t)n * 8;              // N*8
    unsigned* mord  = (unsigned*)(adst + (size_t)n * 8); // N*8
    float*    denom = (float*)mord + (size_t)n * 8;      // N*8
    float*    pooled= denom + (size_t)n * 8;             // 64*16
    float*    cnt   = pooled + N_GRAPHS * 16;            // 64
    float*    out2  = h1;                                // N*128 reuse

    const int TB = 256;
    const long long EH = (long long)(ne + n) * 8;
    const int gridE = (int)((EH + TB - 1) / TB);
    const int gridN = (n + TB - 1) / TB;
    const int gridN64 = (int)(((long long)n * 64 + TB - 1) / TB);

    // pooled + cnt zero
    k_zero<<<(N_GRAPHS * 16 + N_GRAPHS + TB - 1) / TB, TB, 0, stream>>>(pooled,
                                                           N_GRAPHS * 16 + N_GRAPHS);
    // ---- Layer 1 ----
    k_l1_feats<<<gridN, TB, 0, stream>>>(x, W1, as1, ad1, h1, asrc, adst, mord, denom,
                                         out1, n);
    k_edge_max<8><<<gridE, TB, 0, stream>>>(src, dst, ne, n, asrc, adst, mord);
    k_edge_acc<8, 8><<<gridE, TB, 0, stream>>>(src, dst, ne, n, asrc, adst, mord, h1,
                                               denom, out1);
    k_fin1<<<gridN64, TB, 0, stream>>>(out1, denom, b1, n);

    // ---- Layer 2: fp32 WMMA GEMM h2 = h1_act[N,64] @ W2[64,128] ----
    {
        const int strips = (n + 15) / 16;                // one wave per 16x128 strip
        const int wavesPerBlock = TB / 32;
        const int blocks = (strips + wavesPerBlock - 1) / wavesPerBlock;
        k_gemm_wmma_f32<64, 128><<<blocks, TB, 0, stream>>>(out1, W2, h2, n);
    }
    k_l2_att<<<gridN, TB, 0, stream>>>(h2, as2, ad2, asrc, adst, mord, denom, out2, n);
    k_edge_max<8><<<gridE, TB, 0, stream>>>(src, dst, ne, n, asrc, adst, mord);
    k_edge_acc<8, 16><<<gridE, TB, 0, stream>>>(src, dst, ne, n, asrc, adst, mord, h2,
                                                denom, out2);
    k_fin2<<<gridN, TB, 0, stream>>>(out2, denom, b2, batch, pooled, cnt, n);

    // ---- Pool -> FC ----
    k_fc<<<1, TB, 0, stream>>>(pooled, cnt, Wfc, bfc, (float*)d_out);
}